// SymmetricContractionFlax_72894184948352
// MI455X (gfx1250) — compile-verified
//
#include <hip/hip_runtime.h>
#include <stddef.h>

// ---------------------------------------------------------------------------
// Symmetric contraction (MACE-style) for gfx1250 via monomial-basis GEMM on
// V_WMMA_F32_16X16X4_F32.
//
//   out[n,c,i] = sum_o sum_k w_o[s,k,c] * sum_mono u_o[mono,k,i] * P_o[n,c,mono]
// with P_3 = x_a x_b x_j (4096 monomials), P_2 = x_a x_b (256), P_1 = x_a (16).
//
// GEMM: rows = (n,c) (node-major so species is uniform per workgroup),
//       B = u tensors (8.8 MB, L2-resident, pre-swizzled to WMMA B layout),
//       A = monomials generated in-register from x[n,c,0..15].
// ---------------------------------------------------------------------------

typedef float v2f __attribute__((ext_vector_type(2)));
typedef float v8f __attribute__((ext_vector_type(8)));

#if defined(__has_builtin)
#if __has_builtin(__builtin_amdgcn_wmma_f32_16x16x4_f32)
#define HAVE_WMMA_F32X4 1
#endif
#endif

struct WPtrs { const float* w[12]; };   // ws[(order-1)*4 + ir], shape (NS, mul, C)

__device__ __forceinline__ v8f v8zero() {
  v8f r;
#pragma unroll
  for (int i = 0; i < 8; ++i) r[i] = 0.0f;
  return r;
}

// D = A(16x4) * B(4x16) + C(16x16), fp32.
// A layout: lane L, vgpr v -> A[M = L&15][K = 2*(L>>4)+v]
// B layout: lane L, vgpr v -> B[K = 2*(L>>4)+v][N = L&15]
// C layout: lane L, vgpr v -> C[M = v + 8*(L>>4)][N = L&15]
__device__ __forceinline__ v8f wmma4(v2f A, v2f B, v8f C) {
#ifdef HAVE_WMMA_F32X4
  return __builtin_amdgcn_wmma_f32_16x16x4_f32(false, A, false, B, (short)0, C,
                                               false, false);
#else
  // Exact shuffle-based emulation with identical register layout (fallback).
  const int lane = (int)(threadIdx.x & 31);
  const int half = lane >> 4;
  const int N = lane & 15;
#pragma unroll
  for (int v = 0; v < 8; ++v) {
    const int M = v + (half << 3);
    float c = C[v];
#pragma unroll
    for (int k = 0; k < 4; ++k) {
      float a = __shfl((k & 1) ? A.y : A.x, M + ((k >> 1) << 4), 32);
      float b = __shfl((k & 1) ? B.y : B.x, N + ((k >> 1) << 4), 32);
      c = fmaf(a, b, c);
    }
    C[v] = c;
  }
  return C;
#endif
}

// Column -> (ir, k, i) mapping + weight application into LDS out buffer.
template <int ORDER>
__device__ __forceinline__ void epilogue(const v8f& acc, int colBase, int s,
                                         int Mbase, int lane, float* lout,
                                         const WPtrs& wp) {
  constexpr int bnd3[4] = {10, 79, 244, 524};
  constexpr int bnd2[4] = {4, 22, 62, 118};
  constexpr int bnd1[4] = {1, 4, 9, 16};
  constexpr int mm3[4] = {10, 23, 33, 40};
  constexpr int mm2[4] = {4, 6, 8, 8};
  constexpr int mm1[4] = {1, 1, 1, 1};
  const int* bnd = (ORDER == 3) ? bnd3 : ((ORDER == 2) ? bnd2 : bnd1);
  const int* mm  = (ORDER == 3) ? mm3 : ((ORDER == 2) ? mm2 : mm1);
  constexpr int irdim[4] = {1, 3, 5, 7};
  constexpr int iroff[4] = {0, 1, 4, 9};

  const int g = colBase + (lane & 15);
  int base = 0;
#pragma unroll
  for (int ir = 0; ir < 4; ++ir) {
    if (g < bnd[ir]) {
      const int t = g - base;
      const int k = t / irdim[ir];
      const int i = t - k * irdim[ir];
      const float* w = wp.w[(ORDER - 1) * 4 + ir];
      const int widx = (s * mm[ir] + k) * 128;
      const int ioff = iroff[ir] + i;
#pragma unroll
      for (int v = 0; v < 8; ++v) {
        const int c = Mbase + ((lane >> 4) << 3) + v;
        atomicAdd(&lout[c * 16 + ioff], w[widx + c] * acc[v]);
      }
      return;
    }
    base = bnd[ir];
  }
  // g >= bnd[3]: padding column, nothing to do.
}

// One group of NCT (compile-time) column tiles: full K sweep + epilogue.
// No runtime guards -> clean load-clause + wmma inner loop.
template <int ORDER, int NCT, int KSTEPS>
__device__ __forceinline__ void do_group(const float* __restrict__ bp0,
                                         int ctBase, const float* lxc,
                                         const float (&xkA)[4],
                                         const float (&xkB)[4], int lane,
                                         int s, int Mbase, float* lout,
                                         const WPtrs& wp) {
  v8f acc[NCT];
#pragma unroll
  for (int g = 0; g < NCT; ++g) acc[g] = v8zero();
  const float* bp = bp0 + (lane << 1);

  // 4 consecutive K-steps (16 K-values) for all NCT tiles:
  // batch the loads first (one clause), then issue the wmmas.
  auto block4 = [&](int ksbase, float m) {
    v2f Bf[4][NCT];
#pragma unroll
    for (int q = 0; q < 4; ++q)
#pragma unroll
      for (int g = 0; g < NCT; ++g)
        Bf[q][g] = *(const v2f*)(bp + ((size_t)g * KSTEPS + ksbase + q) * 64);
#pragma unroll
    for (int q = 0; q < 4; ++q) {
      v2f A;
      A.x = m * xkA[q];
      A.y = m * xkB[q];
#pragma unroll
      for (int g = 0; g < NCT; ++g) acc[g] = wmma4(A, Bf[q][g], acc[g]);
    }
  };

  if (ORDER == 3) {
    for (int a = 0; a < 16; ++a) {
      const float xa = lxc[a];
      float xb = lxc[0];
      for (int b = 0; b < 16; ++b) {
        const float xbn = lxc[(b + 1) & 15];  // software-pipelined LDS load
        block4((a * 16 + b) * 4, xa * xb);
        xb = xbn;
      }
    }
  } else if (ORDER == 2) {
    for (int a = 0; a < 16; ++a) block4(a * 4, lxc[a]);
  } else {
    block4(0, 1.0f);
  }

#pragma unroll
  for (int g = 0; g < NCT; ++g)
    epilogue<ORDER>(acc[g], (ctBase + g) * 16, s, Mbase, lane, lout, wp);
}

template <int ORDER, int NTILES, int KSTEPS>
__device__ __forceinline__ void run_order(const float* __restrict__ Bs,
                                          const float* lxc,
                                          const float (&xkA)[4],
                                          const float (&xkB)[4], int lane,
                                          int s, int Mbase, float* lout,
                                          const WPtrs& wp) {
  constexpr int NFULL = NTILES / 4;   // groups of 4 tiles
  constexpr int REM = NTILES % 4;     // remainder tiles (0 or 1 here)
  for (int gi = 0; gi < NFULL; ++gi)
    do_group<ORDER, 4, KSTEPS>(Bs + (size_t)gi * 4 * KSTEPS * 64, gi * 4, lxc,
                               xkA, xkB, lane, s, Mbase, lout, wp);
  if constexpr (REM > 0)
    do_group<ORDER, REM, KSTEPS>(Bs + (size_t)NFULL * 4 * KSTEPS * 64,
                                 NFULL * 4, lxc, xkA, xkB, lane, s, Mbase,
                                 lout, wp);
}

__global__ __launch_bounds__(256) void symcon_main(
    const float* __restrict__ x, const int* __restrict__ species, WPtrs wp,
    const float* __restrict__ B3s, const float* __restrict__ B2s,
    const float* __restrict__ B1s, float* __restrict__ out) {
  __shared__ float lx[128 * 17];   // x rows, stride 17 (bank-conflict free)
  __shared__ float lout[2048];     // out accumulator for this node

  const int n = blockIdx.x;
  const int tid = (int)threadIdx.x;
  const int lane = tid & 31;
  const int wave = tid >> 5;
  const int s = species[n];

  for (int t = tid; t < 2048; t += 256) {
    const int c = t >> 4, j = t & 15;
    lx[c * 17 + j] = x[(size_t)n * 2048 + t];
    lout[t] = 0.0f;
  }
  __syncthreads();

  const int Mbase = wave << 4;                    // this wave's channel base
  const int khalf = lane >> 4;                    // K half (lanes 16-31: K=2,3)
  const float* lxc = &lx[(Mbase + (lane & 15)) * 17];

  // Pre-selected per-lane j-values for each K-step: K = 4*q + 2*khalf + {0,1}
  float xkA[4], xkB[4];
#pragma unroll
  for (int q = 0; q < 4; ++q) {
    xkA[q] = lxc[4 * q + 2 * khalf];
    xkB[q] = lxc[4 * q + 2 * khalf + 1];
  }

  run_order<3, 33, 1024>(B3s, lxc, xkA, xkB, lane, s, Mbase, lout, wp);
  run_order<2, 8, 64>(B2s, lxc, xkA, xkB, lane, s, Mbase, lout, wp);
  run_order<1, 1, 4>(B1s, lxc, xkA, xkB, lane, s, Mbase, lout, wp);

  __syncthreads();
  for (int t = tid; t < 2048; t += 256) out[(size_t)n * 2048 + t] = lout[t];
}

// Pre-swizzle u tensors into WMMA B-fragment layout:
//   Bs[((ct*KSTEPS + ks)*32 + lane)*2 + v] = u[K = 4*ks + 2*(lane>>4)+v][col = 16*ct + (lane&15)]
__global__ void build_B_kernel(const float* u0, const float* u1,
                               const float* u2, const float* u3, float* Bs,
                               int KSTEPS, int total, int b0, int b1, int b2,
                               int b3, int m0, int m1, int m2, int m3) {
  const int t = blockIdx.x * blockDim.x + threadIdx.x;
  if (t >= total) return;
  const int v = t & 1;
  const int lane = (t >> 1) & 31;
  const int rem = t >> 6;
  const int ks = rem % KSTEPS;
  const int ct = rem / KSTEPS;
  const int K = ks * 4 + ((lane >> 4) << 1) + v;
  const int col = ct * 16 + (lane & 15);

  const int bnd[4] = {b0, b1, b2, b3};
  const int mm[4] = {m0, m1, m2, m3};
  const float* uu[4] = {u0, u1, u2, u3};
  const int irdim[4] = {1, 3, 5, 7};

  float val = 0.0f;
  int base = 0;
#pragma unroll
  for (int ir = 0; ir < 4; ++ir) {
    if (col < bnd[ir]) {
      const int tt = col - base;
      const int k = tt / irdim[ir];
      const int i = tt - k * irdim[ir];
      val = uu[ir][(size_t)K * (mm[ir] * irdim[ir]) + k * irdim[ir] + i];
      break;
    }
    base = bnd[ir];
  }
  Bs[t] = val;
}

extern "C" void kernel_launch(void* const* d_in, const int* in_sizes, int n_in,
                              void* d_out, int out_size, void* d_ws,
                              size_t ws_size, hipStream_t stream) {
  (void)in_sizes; (void)n_in; (void)out_size;

  const float* x = (const float*)d_in[0];
  const int* index = (const int*)d_in[1];
  WPtrs wp;
  for (int i = 0; i < 12; ++i) wp.w[i] = (const float*)d_in[2 + i];
  const float* u1p[4] = {(const float*)d_in[14], (const float*)d_in[15],
                         (const float*)d_in[16], (const float*)d_in[17]};
  const float* u2p[4] = {(const float*)d_in[18], (const float*)d_in[19],
                         (const float*)d_in[20], (const float*)d_in[21]};
  const float* u3p[4] = {(const float*)d_in[22], (const float*)d_in[23],
                         (const float*)d_in[24], (const float*)d_in[25]};

  const size_t n3 = (size_t)33 * 1024 * 64;  // 2,162,688 floats
  const size_t n2 = (size_t)8 * 64 * 64;     // 32,768 floats
  const size_t n1 = (size_t)1 * 4 * 64;      // 256 floats
  if (ws_size < (n3 + n2 + n1) * sizeof(float)) return;  // scratch too small

  float* B3s = (float*)d_ws;
  float* B2s = B3s + n3;
  float* B1s = B2s + n2;

  build_B_kernel<<<(int)((n3 + 255) / 256), 256, 0, stream>>>(
      u3p[0], u3p[1], u3p[2], u3p[3], B3s, 1024, (int)n3, 10, 79, 244, 524, 10,
      23, 33, 40);
  build_B_kernel<<<(int)((n2 + 255) / 256), 256, 0, stream>>>(
      u2p[0], u2p[1], u2p[2], u2p[3], B2s, 64, (int)n2, 4, 22, 62, 118, 4, 6,
      8, 8);
  build_B_kernel<<<1, 256, 0, stream>>>(u1p[0], u1p[1], u1p[2], u1p[3], B1s, 4,
                                        (int)n1, 1, 4, 9, 16, 1, 1, 1, 1);

  symcon_main<<<512, 256, 0, stream>>>(x, index, wp, B3s, B2s, B1s,
                                       (float*)d_out);
}